// MedVidNet_multi_attn_conv3______10359461118397
// MI455X (gfx1250) — compile-verified
//
#include <hip/hip_runtime.h>
#include <hip/hip_bf16.h>
#include <math.h>

// ---------------- problem constants (from reference) ----------------
#define MAX_F   512
#define BATCH   32
#define DIM     1024
#define NHEAD   16
#define SSUB    64          // DIM / NHEAD
#define NOUT    4
#define FCHID   4096
#define DCAT    4096        // DIM * NOUT
#define MASKV   (-50.0f)

typedef __attribute__((ext_vector_type(2))) float v2f;
typedef __attribute__((ext_vector_type(8))) float v8f;

// ---------------- block reduction helpers (wave32) ----------------
__device__ __forceinline__ float waveReduceSum(float v) {
#pragma unroll
    for (int off = 16; off > 0; off >>= 1) v += __shfl_xor(v, off, 32);
    return v;
}
__device__ __forceinline__ float waveReduceMax(float v) {
#pragma unroll
    for (int off = 16; off > 0; off >>= 1) v = fmaxf(v, __shfl_xor(v, off, 32));
    return v;
}

__device__ __forceinline__ float blockReduceSum(float v, float* red) {
    int wave = threadIdx.x >> 5, lane = threadIdx.x & 31;
    v = waveReduceSum(v);
    if (lane == 0) red[wave] = v;
    __syncthreads();
    if (wave == 0) {
        float x = (lane < 8) ? red[lane] : 0.0f;
        x = waveReduceSum(x);
        if (lane == 0) red[0] = x;
    }
    __syncthreads();
    float r = red[0];
    __syncthreads();
    return r;
}
__device__ __forceinline__ float blockReduceMax(float v, float* red) {
    int wave = threadIdx.x >> 5, lane = threadIdx.x & 31;
    v = waveReduceMax(v);
    if (lane == 0) red[wave] = v;
    __syncthreads();
    if (wave == 0) {
        float x = (lane < 8) ? red[lane] : -1e30f;
        x = waveReduceMax(x);
        if (lane == 0) red[0] = x;
    }
    __syncthreads();
    float r = red[0];
    __syncthreads();
    return r;
}

// =====================================================================
// Kernel 1: per-(b,h) attention. 512 blocks x 256 threads (8 waves).
//   A) alpha[o,f] = <x[f,b,h,:], query[o,h,:]> / 8
//   B) masked softmax over f; write attn to d_out (O,F,B,H)
//   C) h_cat[b, o*1024 + h*64 + s] = sum_{f<nf} attn[o,f] * x[f,b,h,s]
// =====================================================================
__global__ __launch_bounds__(256)
void attn_kernel(const float* __restrict__ x,
                 const int*   __restrict__ num_frames,
                 const float* __restrict__ query,
                 float* __restrict__ hcat,
                 float* __restrict__ attn_out) {
    __shared__ float alpha_s[NOUT * MAX_F];   // 8 KB
    __shared__ float q_s[NOUT * SSUB];        // 1 KB
    __shared__ float red[8];

    const int b = blockIdx.x >> 4;            // /NHEAD
    const int h = blockIdx.x & 15;            // %NHEAD
    const int t = threadIdx.x;
    const int wave = t >> 5, lane = t & 31;
    const int nf = num_frames[b];

    // load 4 query subvectors for this head
    {
        int o = t >> 6, s = t & 63;           // 256 threads == NOUT*SSUB
        q_s[o * SSUB + s] = query[(o * NHEAD + h) * SSUB + s];
    }
    __syncthreads();

    // ---- Phase A: QK dots, one frame per wave per iteration ----
    const float* xbh = x + (size_t)b * DIM + (size_t)h * SSUB;
    for (int f = wave; f < MAX_F; f += 8) {
        const float* xr = xbh + (size_t)f * (BATCH * DIM);
        v2f xv = *(const v2f*)(xr + 2 * lane);       // s = 2*lane, 2*lane+1
        float p0 = xv.x * q_s[0 * SSUB + 2 * lane] + xv.y * q_s[0 * SSUB + 2 * lane + 1];
        float p1 = xv.x * q_s[1 * SSUB + 2 * lane] + xv.y * q_s[1 * SSUB + 2 * lane + 1];
        float p2 = xv.x * q_s[2 * SSUB + 2 * lane] + xv.y * q_s[2 * SSUB + 2 * lane + 1];
        float p3 = xv.x * q_s[3 * SSUB + 2 * lane] + xv.y * q_s[3 * SSUB + 2 * lane + 1];
        p0 = waveReduceSum(p0); p1 = waveReduceSum(p1);
        p2 = waveReduceSum(p2); p3 = waveReduceSum(p3);
        if (lane == 0) {
            alpha_s[0 * MAX_F + f] = p0 * 0.125f;
            alpha_s[1 * MAX_F + f] = p1 * 0.125f;
            alpha_s[2 * MAX_F + f] = p2 * 0.125f;
            alpha_s[3 * MAX_F + f] = p3 * 0.125f;
        }
    }
    __syncthreads();

    // ---- Phase B: masked softmax over frames, per output query ----
    for (int o = 0; o < NOUT; ++o) {
        const int f0 = t, f1 = t + 256;
        float v0 = (f0 < nf) ? alpha_s[o * MAX_F + f0] : MASKV;
        float v1 = (f1 < nf) ? alpha_s[o * MAX_F + f1] : MASKV;
        float m = blockReduceMax(fmaxf(v0, v1), red);
        float e0 = expf(v0 - m), e1 = expf(v1 - m);
        float ssum = blockReduceSum(e0 + e1, red);
        float inv = 1.0f / ssum;
        float a0 = e0 * inv, a1 = e1 * inv;
        alpha_s[o * MAX_F + f0] = a0;
        alpha_s[o * MAX_F + f1] = a1;
        // attn layout (O, F, B, H)
        attn_out[(((size_t)o * MAX_F + f0) * BATCH + b) * NHEAD + h] = a0;
        attn_out[(((size_t)o * MAX_F + f1) * BATCH + b) * NHEAD + h] = a1;
    }
    __syncthreads();

    // ---- Phase C: attn-weighted feature sum (only f < nf contribute) ----
    {
        int o = t >> 6, s = t & 63;
        const float* xp = xbh + s;
        const float* aw = alpha_s + o * MAX_F;
        float acc = 0.0f;
        for (int f = 0; f < nf; ++f)
            acc += aw[f] * xp[(size_t)f * (BATCH * DIM)];
        // h_cat[b, o*H*S + h*S + s]
        hcat[(size_t)b * DCAT + o * (NHEAD * SSUB) + h * SSUB + s] = acc;
    }
}

// =====================================================================
// Kernel 2: Y = relu(h_cat(32x4096) @ W1(4096x4096) + b1)
// fp32 WMMA 16x16x4, one N-tile per wave, both M-tiles per wave so each
// W1 column block streams exactly once. 64 blocks x 128 threads (4 waves).
// =====================================================================
__global__ __launch_bounds__(128)
void fc1_wmma_kernel(const float* __restrict__ hcat,
                     const float* __restrict__ W1,
                     const float* __restrict__ b1,
                     float* __restrict__ Y) {
    const int wave = threadIdx.x >> 5;
    const int lane = threadIdx.x & 31;
    const int ntile = blockIdx.x * 4 + wave;     // 0..255
    const int n0 = ntile * 16;

    const int r16  = lane & 15;                  // row/col within half-wave
    const int kgrp = lane >> 4;                  // 0: K={0,1}, 1: K={2,3}

    v8f c0 = {};                                 // M = 0..15
    v8f c1 = {};                                 // M = 16..31

    // per-lane base pointers
    const float* a0p = hcat + (size_t)r16 * DCAT        + 2 * kgrp;  // A tile m=0..15
    const float* a1p = hcat + (size_t)(16 + r16) * DCAT + 2 * kgrp;  // A tile m=16..31
    const float* bp  = W1 + (size_t)(2 * kgrp) * FCHID + n0 + r16;   // B frag K rows

    for (int kk = 0; kk < DCAT; kk += 4) {
        // A fragments: lane holds A[m][kk+2*kgrp .. +1] (consecutive -> v2f)
        v2f a0 = *(const v2f*)(a0p + kk);
        v2f a1 = *(const v2f*)(a1p + kk);
        // B fragment: lane holds W1[kk+2*kgrp][n], W1[kk+2*kgrp+1][n]
        v2f bf;
        bf.x = bp[(size_t)kk * FCHID];
        bf.y = bp[(size_t)kk * FCHID + FCHID];
        c0 = __builtin_amdgcn_wmma_f32_16x16x4_f32(false, a0, false, bf,
                                                   (short)0, c0, false, false);
        c1 = __builtin_amdgcn_wmma_f32_16x16x4_f32(false, a1, false, bf,
                                                   (short)0, c1, false, false);
    }

    // D layout: VGPR g holds row m = g + 8*(lane>=16), col n = n0 + (lane&15)
    const int nCol  = n0 + r16;
    const int mBase = kgrp * 8;
    const float bias = b1[nCol];
#pragma unroll
    for (int g = 0; g < 8; ++g) {
        int m = mBase + g;
        Y[(size_t)m * FCHID + nCol]        = fmaxf(c0[g] + bias, 0.0f);
        Y[(size_t)(16 + m) * FCHID + nCol] = fmaxf(c1[g] + bias, 0.0f);
    }
}

// =====================================================================
// Kernel 3: out(32x4) = Y(32x4096) @ W2(4096x4) + b2. 32 blocks x 256 thr.
// =====================================================================
__global__ __launch_bounds__(256)
void fc2_kernel(const float* __restrict__ Y,
                const float* __restrict__ W2,
                const float* __restrict__ b2,
                float* __restrict__ out) {
    __shared__ float red[8][4];
    const int b = blockIdx.x;
    const int wave = threadIdx.x >> 5, lane = threadIdx.x & 31;
    float p0 = 0.f, p1 = 0.f, p2 = 0.f, p3 = 0.f;
    const float* yr = Y + (size_t)b * FCHID;
    for (int k = threadIdx.x; k < FCHID; k += 256) {
        float y = yr[k];
        const float* w = W2 + (size_t)k * NOUT;
        p0 += y * w[0]; p1 += y * w[1]; p2 += y * w[2]; p3 += y * w[3];
    }
    p0 = waveReduceSum(p0); p1 = waveReduceSum(p1);
    p2 = waveReduceSum(p2); p3 = waveReduceSum(p3);
    if (lane == 0) { red[wave][0] = p0; red[wave][1] = p1; red[wave][2] = p2; red[wave][3] = p3; }
    __syncthreads();
    if (threadIdx.x < NOUT) {
        float s = b2[threadIdx.x];
#pragma unroll
        for (int w = 0; w < 8; ++w) s += red[w][threadIdx.x];
        out[b * NOUT + threadIdx.x] = s;
    }
}

// =====================================================================
extern "C" void kernel_launch(void* const* d_in, const int* in_sizes, int n_in,
                              void* d_out, int out_size, void* d_ws, size_t ws_size,
                              hipStream_t stream) {
    const float* x          = (const float*)d_in[0];
    const int*   num_frames = (const int*)  d_in[1];
    const float* query      = (const float*)d_in[2];
    const float* W1         = (const float*)d_in[3];
    const float* b1         = (const float*)d_in[4];
    const float* W2         = (const float*)d_in[5];
    const float* b2         = (const float*)d_in[6];

    float* out      = (float*)d_out;            // (32, 4)
    float* attn_out = out + BATCH * NOUT;       // (4, 512, 32, 16)

    float* hcat = (float*)d_ws;                 // (32, 4096)  512 KB
    float* Yact = hcat + (size_t)BATCH * DCAT;  // (32, 4096)  512 KB

    attn_kernel<<<BATCH * NHEAD, 256, 0, stream>>>(x, num_frames, query, hcat, attn_out);
    fc1_wmma_kernel<<<FCHID / 16 / 4, 128, 0, stream>>>(hcat, W1, b1, Yact);
    fc2_kernel<<<BATCH, 256, 0, stream>>>(Yact, W2, b2, out);
}